// NCL_50766513438744
// MI455X (gfx1250) — compile-verified
//
#include <hip/hip_runtime.h>

typedef __attribute__((ext_vector_type(2))) float v2f;
typedef __attribute__((ext_vector_type(8))) float v8f;

// D = A(16x4 f32) * B(4x16 f32) + C(16x16 f32), full fp32 CDNA5 WMMA
#define WMMA4(a, b, c) \
  __builtin_amdgcn_wmma_f32_16x16x4_f32(false, (a), false, (b), (short)0, (c), false, false)

#define NWAVES 4
#define FSTR   34                    // padded feature stride (floats), bank-conflict free
#define ACT_F  (4 * 16 * FSTR)       // per-wave staging: 4 channels x 16 samples x 34
#define JV_F   (3 * 16 * 4)          // per-wave Jacobian staging

// Raw-hardware softplus(20x)/20 and sigmoid(20x), sharing one v_exp_f32.
// All arguments are in-range for the raw ops: d = 1+e in [1,2] (no denorm/inf),
// exp2 of a large negative underflows cleanly to 0.
//   e = 2^(-20*log2e*|x|) = exp(-|20x|)
//   softplus(20x)/20 = max(x,0) + log2(1+e) * (ln2/20)
//   sigmoid(20x)     = x>=0 ? 1/(1+e) : e/(1+e)
#define NEG20_LOG2E  (-28.853900817779268f)   // -20 / ln(2)
#define LN2_OVER_20  (0.034657359027997264f)  // ln(2) / 20

__device__ __forceinline__ void act_pair(float pre, float& h, float& s) {
  float e = __builtin_amdgcn_exp2f(NEG20_LOG2E * __builtin_fabsf(pre));
  float d = 1.0f + e;
  float r = __builtin_amdgcn_rcpf(d);               // v_rcp_f32
  float l = __builtin_amdgcn_logf(d);               // v_log_f32 (log2), d in [1,2]
  h = fmaxf(pre, 0.0f) + l * LN2_OVER_20;
  s = (pre >= 0.0f) ? r : e * r;
}
__device__ __forceinline__ float sig_only(float pre) {
  float e = __builtin_amdgcn_exp2f(NEG20_LOG2E * __builtin_fabsf(pre));
  float r = __builtin_amdgcn_rcpf(1.0f + e);
  return (pre >= 0.0f) ? r : e * r;
}

// store a 16x16 D tile (M=r+8g samples, N=lane col) into [channel][sample][feat]
__device__ __forceinline__ void store_d(float* act, int c, int nt, int g, int n16, v8f d) {
#pragma unroll
  for (int r = 0; r < 8; ++r)
    act[(c * 16 + r + 8 * g) * FSTR + nt * 16 + n16] = d[r];
}

// load A-operand pair for K-step kk: lane holds A[M=n16][K = kk*4 + 2g + {0,1}]
__device__ __forceinline__ v2f load_a2(const float* act, int c, int kk, int g, int n16) {
  const float* p = act + (c * 16 + n16) * FSTR + kk * 4 + 2 * g;
  return *(const v2f*)p;  // 8B aligned: FSTR even, offsets even
}

// one hidden layer (32->32) for forward + 3 tangent channels, staged via LDS.
// NEED_H=false skips storing the forward softplus value (layer 3: only s3 is used).
template <bool NEED_H>
__device__ __forceinline__ void hidden_layer(float* act, const v2f (&WB)[2][8],
                                             const float (&bv)[2], int g, int n16) {
  v8f sall0, sall1;
  {
    v8f acc0, acc1;
#pragma unroll
    for (int r = 0; r < 8; ++r) { acc0[r] = bv[0]; acc1[r] = bv[1]; }
#pragma unroll
    for (int kk = 0; kk < 8; ++kk) {
      v2f a = load_a2(act, 0, kk, g, n16);
      acc0 = WMMA4(a, WB[0][kk], acc0);
      acc1 = WMMA4(a, WB[1][kk], acc1);
    }
    if (NEED_H) {
      v8f h0, h1;
#pragma unroll
      for (int r = 0; r < 8; ++r) {
        float hh, ss;
        act_pair(acc0[r], hh, ss); h0[r] = hh; sall0[r] = ss;
        act_pair(acc1[r], hh, ss); h1[r] = hh; sall1[r] = ss;
      }
      store_d(act, 0, 0, g, n16, h0);
      store_d(act, 0, 1, g, n16, h1);
    } else {
#pragma unroll
      for (int r = 0; r < 8; ++r) {
        sall0[r] = sig_only(acc0[r]);
        sall1[r] = sig_only(acc1[r]);
      }
    }
  }
#pragma unroll
  for (int i = 1; i <= 3; ++i) {
    v8f acc0, acc1;
#pragma unroll
    for (int r = 0; r < 8; ++r) { acc0[r] = 0.0f; acc1[r] = 0.0f; }
#pragma unroll
    for (int kk = 0; kk < 8; ++kk) {
      v2f a = load_a2(act, i, kk, g, n16);
      acc0 = WMMA4(a, WB[0][kk], acc0);
      acc1 = WMMA4(a, WB[1][kk], acc1);
    }
    acc0 *= sall0;
    acc1 *= sall1;
    store_d(act, i, 0, g, n16, acc0);
    store_d(act, i, 1, g, n16, acc1);
  }
}

__global__ __launch_bounds__(128) void fused_mlp_jac_trace(
    const float* __restrict__ x,
    const float* __restrict__ W1, const float* __restrict__ b1,
    const float* __restrict__ W2, const float* __restrict__ b2,
    const float* __restrict__ W3, const float* __restrict__ b3,
    const float* __restrict__ W4,
    float* __restrict__ out, int N, int ntiles) {
  __shared__ float lds_act[NWAVES][ACT_F];
  __shared__ float lds_jv[NWAVES][JV_F];

  const int lane = threadIdx.x & 31;
  const int wave = threadIdx.x >> 5;
  const int g    = lane >> 4;   // lane group (0/1)
  const int n16  = lane & 15;

  float* act = lds_act[wave];
  float* jvb = lds_jv[wave];

  // ---- preload weights into WMMA B-operand layout: B[K = kk*4 + v + 2g][N = nt*16 + n16]
  v2f W1B[2];        // single K-step (K=3 padded to 4)
  v2f W2B[2][8];
  v2f W3B[2][8];
  v2f W4B[8];        // N=3 padded to 16 (single N-tile)
#pragma unroll
  for (int nt = 0; nt < 2; ++nt) {
#pragma unroll
    for (int v = 0; v < 2; ++v) {
      int k = v + 2 * g;
      W1B[nt][v] = (k < 3) ? W1[k * 32 + nt * 16 + n16] : 0.0f;
    }
#pragma unroll
    for (int kk = 0; kk < 8; ++kk) {
#pragma unroll
      for (int v = 0; v < 2; ++v) {
        int k = kk * 4 + v + 2 * g;
        W2B[nt][kk][v] = W2[k * 32 + nt * 16 + n16];
        W3B[nt][kk][v] = W3[k * 32 + nt * 16 + n16];
      }
    }
  }
#pragma unroll
  for (int kk = 0; kk < 8; ++kk) {
#pragma unroll
    for (int v = 0; v < 2; ++v) {
      int k = kk * 4 + v + 2 * g;
      W4B[kk][v] = (n16 < 3) ? W4[k * 3 + n16] : 0.0f;
    }
  }
  // per-lane bias and W1 rows (feature = nt*16 + n16) for tangent seeding
  float b1v[2], b2v[2], b3v[2], w1row[3][2];
#pragma unroll
  for (int nt = 0; nt < 2; ++nt) {
    b1v[nt] = b1[nt * 16 + n16];
    b2v[nt] = b2[nt * 16 + n16];
    b3v[nt] = b3[nt * 16 + n16];
#pragma unroll
    for (int i = 0; i < 3; ++i) w1row[i][nt] = W1[i * 32 + nt * 16 + n16];
  }

  const int waveId = blockIdx.x * NWAVES + wave;
  const int nw     = gridDim.x * NWAVES;

  for (int t = waveId; t < ntiles; t += nw) {
    const int s0 = t * 16;

    // ---- layer 1: pre1 = x @ W1 + b1 ; seed tangents T1_i = s1 * W1[i,:]
    int sidx = s0 + n16; if (sidx > N - 1) sidx = N - 1;
    const float* xp = x + (size_t)sidx * 3;
    // unconditional loads + lane-group select (no divergent exec around WMMA)
    float x0 = xp[0], x1 = xp[1], x2 = xp[2];
    v2f a1;
    a1[0] = g ? x2 : x0;
    a1[1] = g ? 0.0f : x1;
#pragma unroll
    for (int nt = 0; nt < 2; ++nt) {
      v8f cin;
#pragma unroll
      for (int r = 0; r < 8; ++r) cin[r] = b1v[nt];
      v8f pre = WMMA4(a1, W1B[nt], cin);
      v8f h, s;
#pragma unroll
      for (int r = 0; r < 8; ++r) {
        float hh, ss;
        act_pair(pre[r], hh, ss);
        h[r] = hh; s[r] = ss;
      }
      store_d(act, 0, nt, g, n16, h);
#pragma unroll
      for (int i = 0; i < 3; ++i) {
        v8f tt;
#pragma unroll
        for (int r = 0; r < 8; ++r) tt[r] = w1row[i][nt] * s[r];
        store_d(act, 1 + i, nt, g, n16, tt);
      }
    }

    // ---- layers 2 and 3 (32 -> 32, forward + 3 tangents)
    hidden_layer<true >(act, W2B, b2v, g, n16);   // h2 feeds layer 3
    hidden_layer<false>(act, W3B, b3v, g, n16);   // only s3 is needed

    // ---- layer 4: Jv_i = T3_i @ W4  (bias drops out of the Jacobian)
#pragma unroll
    for (int i = 0; i < 3; ++i) {
      v8f acc;
#pragma unroll
      for (int r = 0; r < 8; ++r) acc[r] = 0.0f;
#pragma unroll
      for (int kk = 0; kk < 8; ++kk) {
        v2f a = load_a2(act, 1 + i, kk, g, n16);
        acc = WMMA4(a, W4B[kk], acc);
      }
      if (n16 < 3) {
#pragma unroll
        for (int r = 0; r < 8; ++r)
          jvb[(i * 16 + r + 8 * g) * 4 + n16] = acc[r];
      }
    }

    // ---- combine trace terms and write out[n, 0..2]
    if (lane < 16) {
      int so = s0 + lane;
      if (so < N) {
        float j00 = jvb[(0 * 16 + lane) * 4 + 0];
        float j01 = jvb[(0 * 16 + lane) * 4 + 1];
        float j10 = jvb[(1 * 16 + lane) * 4 + 0];
        float j12 = jvb[(1 * 16 + lane) * 4 + 2];
        float j21 = jvb[(2 * 16 + lane) * 4 + 1];
        float j22 = jvb[(2 * 16 + lane) * 4 + 2];
        float* op = out + (size_t)so * 3;
        op[0] = j10 + j21;    // +dv0/dx1 + dv1/dx2
        op[1] = j22 - j00;    // -dv0/dx0 + dv2/dx2
        op[2] = -j01 - j12;   // -dv1/dx0 - dv2/dx1
      }
    }
  }
}

extern "C" void kernel_launch(void* const* d_in, const int* in_sizes, int n_in,
                              void* d_out, int out_size, void* d_ws, size_t ws_size,
                              hipStream_t stream) {
  const float* x  = (const float*)d_in[0];
  const float* W1 = (const float*)d_in[1];
  const float* b1 = (const float*)d_in[2];
  const float* W2 = (const float*)d_in[3];
  const float* b2 = (const float*)d_in[4];
  const float* W3 = (const float*)d_in[5];
  const float* b3 = (const float*)d_in[6];
  const float* W4 = (const float*)d_in[7];
  // d_in[8] (b4) is unused: it drops out of the Jacobian.

  const int N      = in_sizes[0] / 3;
  const int ntiles = (N + 15) / 16;

  int nblocks = (ntiles + NWAVES - 1) / NWAVES;
  if (nblocks > 4096) nblocks = 4096;
  if (nblocks < 1) nblocks = 1;

  fused_mlp_jac_trace<<<nblocks, NWAVES * 32, 0, stream>>>(
      x, W1, b1, W2, b2, W3, b3, W4, (float*)d_out, N, ntiles);
}